// OutputUnit_730144440751
// MI455X (gfx1250) — compile-verified
//
#include <hip/hip_runtime.h>
#include <math.h>

typedef __attribute__((ext_vector_type(2))) float v2f;
typedef __attribute__((ext_vector_type(4))) float f32x4;
typedef __attribute__((ext_vector_type(8))) float v8f;

#define B_SZ 256
#define D_SZ 512
#define C_SZ 1024
#define H_SZ 768
#define MASK_VAL (-1000.0f)

// ---------------------------------------------------------------------------
// Stage 1: h = ELU(concat(mp, q) @ W1 + b1)     [256 x 512], K = 1024
// One wave (32 lanes) computes one 16x16 tile via V_WMMA_F32_16X16X4_F32.
// ---------------------------------------------------------------------------
__global__ __launch_bounds__(32) void mlp1_kernel(
    const float* __restrict__ mp, const float* __restrict__ q,
    const float* __restrict__ W1, const float* __restrict__ b1,
    float* __restrict__ h_out)
{
    const int NT = D_SZ / 16;              // 32 N-tiles
    int tile = blockIdx.x;                 // 16 * 32 = 512 tiles
    int mt = tile / NT;
    int nt = tile % NT;
    int lane = threadIdx.x;
    int m  = mt * 16 + (lane & 15);        // A row for this lane
    int n  = nt * 16 + (lane & 15);        // B col for this lane
    int kh = (lane >> 4) * 2;              // K sub-offset: 0 or 2

    v8f acc = {};
    for (int k0 = 0; k0 < 2 * D_SZ; k0 += 4) {
        int ka = k0 + kh;                  // k0 % 4 == 0, so (ka, ka+1) never straddle 512
        const float* asrc = (ka < D_SZ) ? (mp + (size_t)m * D_SZ + ka)
                                        : (q  + (size_t)m * D_SZ + (ka - D_SZ));
        v2f a; a.x = asrc[0]; a.y = asrc[1];
        v2f b;
        b.x = W1[(size_t)(ka + 0) * D_SZ + n];
        b.y = W1[(size_t)(ka + 1) * D_SZ + n];
        acc = __builtin_amdgcn_wmma_f32_16x16x4_f32(
            false, a, false, b, (short)0, acc, false, false);
    }

    float bias = b1[n];
    int mbase = mt * 16 + ((lane >> 4) * 8);
#pragma unroll
    for (int v = 0; v < 8; ++v) {
        float x = acc[v] + bias;
        x = (x > 0.0f) ? x : (__expf(x) - 1.0f);   // ELU
        h_out[(size_t)(mbase + v) * D_SZ + n] = x;
    }
}

// ---------------------------------------------------------------------------
// Stage 2: o = h @ W2 + b2                      [256 x 768], K = 512
// ---------------------------------------------------------------------------
__global__ __launch_bounds__(32) void mlp2_kernel(
    const float* __restrict__ h, const float* __restrict__ W2,
    const float* __restrict__ b2, float* __restrict__ o_out)
{
    const int NT = H_SZ / 16;              // 48 N-tiles
    int tile = blockIdx.x;                 // 16 * 48 = 768 tiles
    int mt = tile / NT;
    int nt = tile % NT;
    int lane = threadIdx.x;
    int m  = mt * 16 + (lane & 15);
    int n  = nt * 16 + (lane & 15);
    int kh = (lane >> 4) * 2;

    v8f acc = {};
    for (int k0 = 0; k0 < D_SZ; k0 += 4) {
        int ka = k0 + kh;
        v2f a;
        a.x = h[(size_t)m * D_SZ + ka + 0];
        a.y = h[(size_t)m * D_SZ + ka + 1];
        v2f b;
        b.x = W2[(size_t)(ka + 0) * H_SZ + n];
        b.y = W2[(size_t)(ka + 1) * H_SZ + n];
        acc = __builtin_amdgcn_wmma_f32_16x16x4_f32(
            false, a, false, b, (short)0, acc, false, false);
    }

    float bias = b2[n];
    int mbase = mt * 16 + ((lane >> 4) * 8);
#pragma unroll
    for (int v = 0; v < 8; ++v)
        o_out[(size_t)(mbase + v) * H_SZ + n] = acc[v] + bias;
}

// ---------------------------------------------------------------------------
// Stage 3: scores[b,c] = dot(candidates[b,c,:], o[b,:]), ragged-masked.
// 805 MB single-use stream -> 128-bit non-temporal loads.
// o[b] staged LDS -> registers once per wave; inner region is pure
// global_load_b128 + FMA with 4 parallel row accumulators (24 loads in
// flight per wave) to saturate HBM.
// ---------------------------------------------------------------------------
#define ROWS_PER_WAVE 4
#define ROWS_PER_BLOCK (8 * ROWS_PER_WAVE)   // 32
#define HCHUNKS (H_SZ / (4 * 32))            // 6 f32x4 chunks per lane

__global__ __launch_bounds__(256) void score_kernel(
    const float* __restrict__ cand, const float* __restrict__ o,
    const int* __restrict__ counts, float* __restrict__ out)
{
    __shared__ f32x4 s_o[H_SZ / 4];        // 192 x 16B = 3 KB

    int b   = blockIdx.y;
    int tid = threadIdx.x;

    // cooperative load of o[b] into LDS (16B-aligned: H*4 = 3072 bytes)
    const f32x4* o4 = (const f32x4*)(o + (size_t)b * H_SZ);
    if (tid < H_SZ / 4) s_o[tid] = o4[tid];
    __syncthreads();

    int wave = tid >> 5;
    int lane = tid & 31;

    // hoist o[b] slice into registers: invariant across all rows this wave does
    f32x4 w[HCHUNKS];
#pragma unroll
    for (int i = 0; i < HCHUNKS; ++i) w[i] = s_o[lane + i * 32];

    int cbase = blockIdx.x * ROWS_PER_BLOCK + wave * ROWS_PER_WAVE;
    const f32x4* row0 = (const f32x4*)(cand + ((size_t)b * C_SZ + cbase) * H_SZ);

    float sum[ROWS_PER_WAVE];
#pragma unroll
    for (int r = 0; r < ROWS_PER_WAVE; ++r) sum[r] = 0.0f;

    // fully unrolled: 24 NT b128 loads + FMAs, no LDS, no reductions inside
#pragma unroll
    for (int r = 0; r < ROWS_PER_WAVE; ++r) {
        const f32x4* row = row0 + (size_t)r * (H_SZ / 4);
#pragma unroll
        for (int i = 0; i < HCHUNKS; ++i) {
            f32x4 v = __builtin_nontemporal_load(&row[lane + i * 32]);
            sum[r] = fmaf(v.x, w[i].x, sum[r]);
            sum[r] = fmaf(v.y, w[i].y, sum[r]);
            sum[r] = fmaf(v.z, w[i].z, sum[r]);
            sum[r] = fmaf(v.w, w[i].w, sum[r]);
        }
    }

    // wave32 butterfly reductions + masked stores
    int nvalid = counts[b];
#pragma unroll
    for (int r = 0; r < ROWS_PER_WAVE; ++r) {
#pragma unroll
        for (int off = 16; off > 0; off >>= 1)
            sum[r] += __shfl_xor(sum[r], off, 32);
        int c = cbase + r;
        if (lane == 0)
            out[(size_t)b * C_SZ + c] = (c < nvalid) ? sum[r] : MASK_VAL;
    }
}

// ---------------------------------------------------------------------------
extern "C" void kernel_launch(void* const* d_in, const int* in_sizes, int n_in,
                              void* d_out, int out_size, void* d_ws, size_t ws_size,
                              hipStream_t stream) {
    const float* mp     = (const float*)d_in[0];
    const float* q      = (const float*)d_in[1];
    const float* cand   = (const float*)d_in[2];
    const float* W1     = (const float*)d_in[3];
    const float* b1     = (const float*)d_in[4];
    const float* W2     = (const float*)d_in[5];
    const float* b2     = (const float*)d_in[6];
    const int*   counts = (const int*)d_in[7];
    float* out = (float*)d_out;

    float* h = (float*)d_ws;                       // 256*512 floats (512 KB)
    float* o = h + (size_t)B_SZ * D_SZ;            // 256*768 floats (768 KB)

    mlp1_kernel<<<dim3((B_SZ / 16) * (D_SZ / 16)), dim3(32), 0, stream>>>(mp, q, W1, b1, h);
    mlp2_kernel<<<dim3((B_SZ / 16) * (H_SZ / 16)), dim3(32), 0, stream>>>(h, W2, b2, o);
    score_kernel<<<dim3(C_SZ / ROWS_PER_BLOCK, B_SZ), dim3(256), 0, stream>>>(cand, o, counts, out);
}